// GCN_58720792870991
// MI455X (gfx1250) — compile-verified
//
#include <hip/hip_runtime.h>
#include <math.h>

// ---------------------------------------------------------------------------
// GCN forward for MI455X (gfx1250, wave32, WMMA).
//   H0 = X @ W1                      (fp32 WMMA 16x16x4)
//   A1 = scatter_add(val * H0[src])  (global_atomic_add_f32, L2-resident)
//   H2 = relu(A1) @ W2               (fp32 WMMA 16x16x4, relu fused in A load)
//   A2 = scatter_add(val * H2[src])
//   out = log_softmax(A2)
// Workspace: H0 6.4M | A1 6.4M | H2 0.8M | A2 0.8M floats = 57.6 MB.
// ---------------------------------------------------------------------------

#define N_NODES 50000
#define N_EDGES 800000
#define IN_F    512
#define HID_F   128
#define OUT_F   16
#define ROW_TILES (N_NODES / 16)   // 3125, exact

typedef __attribute__((ext_vector_type(2))) float v2f;
typedef __attribute__((ext_vector_type(8))) float v8f;

// ------------------------------- utilities ---------------------------------

__global__ void zero_f32(float* __restrict__ p, int n) {
  int i = blockIdx.x * blockDim.x + threadIdx.x;
  int stride = gridDim.x * blockDim.x;
  for (; i < n; i += stride) p[i] = 0.0f;
}

// ------------------------- GEMM1: X[50000,512] @ W1[512,128] ----------------
// Block = 256 threads = 8 wave32s. Block owns one 16-row tile of X (staged in
// LDS, padded stride 516 -> 16 lanes hit 16 distinct bank-pairs on ds_load_b64).
// Wave w computes the 16x16 tile at columns [16w, 16w+16).
//
// WMMA f32 16x16x4 operand layout (ISA 7.12.2):
//   A: lane L holds A[M = L&15][K = 2*(L>>4) + v] in VGPR v (v = 0,1)
//   B: lane L holds B[K = 2*(L>>4) + v][N = L&15] in VGPR v
//   C/D: lane L, VGPR r holds C[M = 8*(L>>4) + r][N = L&15]

#define LDS_STRIDE 516  // 512 + 4 floats pad: 516 % 64 == 4 -> conflict-free

__global__ void __launch_bounds__(256)
gemm1_wmma(const float* __restrict__ X, const float* __restrict__ W1,
           float* __restrict__ H0) {
  __shared__ float sA[16 * LDS_STRIDE];  // 33,024 bytes

  const int rowTile = blockIdx.x;        // 0..3124
  const int row0 = rowTile * 16;

  // Cooperative stage: 16 rows x 128 float4 = 2048 float4 / 256 threads.
  {
    const int t = threadIdx.x;
#pragma unroll
    for (int i = 0; i < 8; ++i) {
      const int g  = t + i * 256;
      const int r  = g >> 7;      // row 0..15
      const int c4 = g & 127;     // float4 column
      const float4 v = ((const float4*)(X + (size_t)(row0 + r) * IN_F))[c4];
      *(float4*)(sA + r * LDS_STRIDE + c4 * 4) = v;  // 2064B row pitch, 16B aligned
    }
  }
  __syncthreads();

  const int lane = threadIdx.x & 31;
  const int wave = threadIdx.x >> 5;     // 0..7 -> column tile
  const int col0 = wave * 16;
  const int m    = lane & 15;            // M for A, N for B
  const int kh   = (lane >> 4) * 2;      // K sub-offset for this half-wave

  const float* sArow = sA + m * LDS_STRIDE + kh;
  const float* Bcol  = W1 + (size_t)kh * HID_F + (col0 + m);

  v8f acc = {};
#pragma unroll 4
  for (int k = 0; k < IN_F; k += 4) {
    v2f a = *(const v2f*)(sArow + k);              // ds_load_b64
    v2f b;
    b.x = Bcol[(size_t)k * HID_F];                 // global, L2-resident W1
    b.y = Bcol[(size_t)(k + 1) * HID_F];
    acc = __builtin_amdgcn_wmma_f32_16x16x4_f32(
        false, a, false, b, (short)0, acc, false, false);
  }

  float* out = H0 + (size_t)(row0 + (lane >> 4) * 8) * HID_F + col0 + m;
#pragma unroll
  for (int r = 0; r < 8; ++r) out[(size_t)r * HID_F] = acc[r];
}

// ------------------- GEMM2: relu(A1)[50000,128] @ W2[128,16] ----------------
// One wave per 16x16 output tile (only one column tile). ReLU fused into the
// A-operand loads. W2 (8 KB) lives in WGP$/L2.

__global__ void __launch_bounds__(256)
gemm2_wmma(const float* __restrict__ A1, const float* __restrict__ W2,
           float* __restrict__ H2) {
  const int lane = threadIdx.x & 31;
  const int wave = threadIdx.x >> 5;
  const int tile = blockIdx.x * 8 + wave;          // wave-uniform guard
  if (tile >= ROW_TILES) return;

  const int row0 = tile * 16;
  const int m    = lane & 15;
  const int kh   = (lane >> 4) * 2;

  const float* Arow = A1 + (size_t)(row0 + m) * HID_F + kh;  // 8B aligned
  const float* Bcol = W2 + (size_t)kh * OUT_F + m;

  v8f acc = {};
#pragma unroll 8
  for (int k = 0; k < HID_F; k += 4) {
    v2f a = *(const v2f*)(Arow + k);
    a.x = fmaxf(a.x, 0.0f);                        // fused ReLU
    a.y = fmaxf(a.y, 0.0f);
    v2f b;
    b.x = Bcol[(size_t)k * OUT_F];
    b.y = Bcol[(size_t)(k + 1) * OUT_F];
    acc = __builtin_amdgcn_wmma_f32_16x16x4_f32(
        false, a, false, b, (short)0, acc, false, false);
  }

  float* out = H2 + (size_t)(row0 + (lane >> 4) * 8) * OUT_F + m;
#pragma unroll
  for (int r = 0; r < 8; ++r) out[(size_t)r * OUT_F] = acc[r];
}

// ------------------------------ SpMM scatter --------------------------------
// One thread per (edge, 4-feature chunk): consecutive threads cover one edge's
// row -> coalesced float4 gather; scatter via no-return global_atomic_add_f32.
// Both source rows and the accumulation buffer fit in the 192 MB L2.

template <int NF4>  // features / 4 (power of two: 32 or 4)
__global__ void __launch_bounds__(256)
spmm_scatter(const float* __restrict__ H, const int* __restrict__ esrc,
             const int* __restrict__ edst, const float* __restrict__ eval,
             float* __restrict__ agg) {
  const int gid = blockIdx.x * blockDim.x + threadIdx.x;  // exact grid
  const int e  = gid >> (NF4 == 32 ? 5 : 2);
  const int fb = gid & (NF4 - 1);
  const int F  = NF4 * 4;

  const int   s = esrc[e];
  const int   d = edst[e];
  const float v = eval[e];

  const float4 h = ((const float4*)(H + (size_t)s * F))[fb];
  float* a = agg + (size_t)d * F + fb * 4;
  unsafeAtomicAdd(a + 0, v * h.x);
  unsafeAtomicAdd(a + 1, v * h.y);
  unsafeAtomicAdd(a + 2, v * h.z);
  unsafeAtomicAdd(a + 3, v * h.w);
}

// ------------------------------ log_softmax ---------------------------------

__global__ void __launch_bounds__(256)
log_softmax16(const float* __restrict__ A, float* __restrict__ out) {
  const int i = blockIdx.x * blockDim.x + threadIdx.x;
  if (i >= N_NODES) return;
  const float4* r = (const float4*)(A + (size_t)i * OUT_F);
  float v[16];
#pragma unroll
  for (int q = 0; q < 4; ++q) {
    const float4 t = r[q];
    v[q * 4 + 0] = t.x; v[q * 4 + 1] = t.y; v[q * 4 + 2] = t.z; v[q * 4 + 3] = t.w;
  }
  float mx = v[0];
#pragma unroll
  for (int q = 1; q < 16; ++q) mx = fmaxf(mx, v[q]);
  float s = 0.0f;
#pragma unroll
  for (int q = 0; q < 16; ++q) s += __expf(v[q] - mx);
  const float lse = mx + __logf(s);
  float* o = out + (size_t)i * OUT_F;
#pragma unroll
  for (int q = 0; q < 16; ++q) o[q] = v[q] - lse;
}

// ------------------------------ launcher ------------------------------------

extern "C" void kernel_launch(void* const* d_in, const int* in_sizes, int n_in,
                              void* d_out, int out_size, void* d_ws, size_t ws_size,
                              hipStream_t stream) {
  (void)in_sizes; (void)n_in; (void)out_size;
  const float* x    = (const float*)d_in[0];
  const int*   esrc = (const int*)d_in[1];
  const int*   edst = (const int*)d_in[2];
  const float* eval = (const float*)d_in[3];
  const float* W1   = (const float*)d_in[4];
  const float* W2   = (const float*)d_in[5];

  float* ws = (float*)d_ws;
  float* H0 = ws;                 // 50000*128 = 6,400,000
  float* A1 = ws + 6400000;       // 6,400,000
  float* H2 = ws + 12800000;      //   800,000
  float* A2 = ws + 13600000;      //   800,000  (total 57.6 MB)
  if (ws_size < (size_t)14400000 * sizeof(float)) return;

  zero_f32<<<2048, 256, 0, stream>>>(A1, 6400000);
  zero_f32<<<512,  256, 0, stream>>>(A2, 800000);

  gemm1_wmma<<<ROW_TILES, 256, 0, stream>>>(x, W1, H0);                 // 3125 blocks
  spmm_scatter<32><<<(N_EDGES * 32) / 256, 256, 0, stream>>>(H0, esrc, edst, eval, A1);
  gemm2_wmma<<<(ROW_TILES + 7) / 8, 256, 0, stream>>>(A1, W2, H2);      // 391 blocks
  spmm_scatter<4><<<(N_EDGES * 4) / 256, 256, 0, stream>>>(H2, esrc, edst, eval, A2);
  log_softmax16<<<(N_NODES + 255) / 256, 256, 0, stream>>>(A2, (float*)d_out);
}